// GCN_63032940036157
// MI455X (gfx1250) — compile-verified
//
#include <hip/hip_runtime.h>
#include <math.h>
#include <stdint.h>

typedef float v2f __attribute__((ext_vector_type(2)));
typedef float v8f __attribute__((ext_vector_type(8)));

#define NFEAT 128
#define LDS_STRIDE 132   // x tile: 128 + 4 pad, 528B rows keep 16B alignment
#define WSTRIDE 132      // swizzled W: per-col stride in floats (2*64 used + 4 pad)

// ---------------- utility: zero a float buffer ----------------
__global__ void zero_f32(float* __restrict__ p, long long n) {
  long long i = (long long)blockIdx.x * blockDim.x + threadIdx.x;
  long long stride = (long long)gridDim.x * blockDim.x;
  for (; i < n; i += stride) p[i] = 0.0f;
}

// ---------------- degree: deg[dst[e]] += 1 ----------------
__global__ void degree_kernel(const int* __restrict__ dst, float* __restrict__ deg, int E) {
  int i = blockIdx.x * blockDim.x + threadIdx.x;
  int stride = gridDim.x * blockDim.x;
  for (; i < E; i += stride) atomicAdd(&deg[dst[i]], 1.0f);
}

// ---------------- dinv = rsqrt(deg + 1)  (self-loop makes deg>0 always) ----------------
__global__ void dinv_kernel(float* __restrict__ deg, int N) {
  int i = blockIdx.x * blockDim.x + threadIdx.x;
  if (i < N) deg[i] = rsqrtf(deg[i] + 1.0f);
}

// Async global -> LDS copy of 16 bytes (GLOBAL_LOAD_ASYNC_TO_LDS_B128, GV mode,
// tracked by ASYNCcnt). lds_off = LDS byte address (low 32 bits of flat shared ptr).
__device__ __forceinline__ void async_copy_b128(uint32_t lds_off, const void* gptr) {
  asm volatile("global_load_async_to_lds_b128 %0, %1, off"
               :: "v"(lds_off), "v"((unsigned long long)(uintptr_t)gptr)
               : "memory");
}
__device__ __forceinline__ void wait_asynccnt0() {
  asm volatile("s_wait_asynccnt 0x0" ::: "memory");
}

// ---------------- WMMA GEMM: xw[m,n] = sum_k x[m,k] * W[n,k] ----------------
// Block: 256 threads = 8 waves, handles 128 rows of x. Wave w owns rows
// [16w, 16w+16) and sweeps 8 column tiles with V_WMMA_F32_16X16X4_F32.
// x tile staged via async global->LDS (ASYNCcnt). W staged re-swizzled into
// fragment order so each lane's B operand for a tile is 64 contiguous floats
// (16 x ds_load_b128). sched_barrier pins loads ahead of the WMMA chain.
__global__ __launch_bounds__(256) void gemm_xw_wmma(
    const float* __restrict__ x, const float* __restrict__ W,
    float* __restrict__ xw, int N) {
  extern __shared__ float smem[];
  float* sX  = smem;                      // [128][LDS_STRIDE], memory order
  float* sWs = smem + 128 * LDS_STRIDE;   // [col][half][k4][2]: col*WSTRIDE + half*64 + k4*2

  const int tid = threadIdx.x;
  const int row_base = blockIdx.x * 128;

  // x tile: async global->LDS b128 (row-clamped at tail)
#pragma unroll
  for (int i = 0; i < 16; ++i) {
    int j  = tid + i * 256;        // float4 index 0..4095
    int r  = j >> 5;               // row 0..127
    int c4 = j & 31;               // float4 column 0..31
    int gr = row_base + r; if (gr >= N) gr = N - 1;
    uint32_t lx = (uint32_t)(uintptr_t)&sX[r * LDS_STRIDE + (c4 << 2)];
    async_copy_b128(lx, x + (size_t)gr * NFEAT + (c4 << 2));
  }
  // W: vmem float4 load, swizzle into fragment order with two b64 LDS stores.
  // float4 at (n, k=4*k4) -> half0 pair (k,k+1) at +2*k4, half1 pair (k+2,k+3) at +64+2*k4.
#pragma unroll
  for (int i = 0; i < 16; ++i) {
    int j  = tid + i * 256;
    int n  = j >> 5;               // output feature (W row) 0..127
    int k4 = j & 31;               // k/4
    float4 w = ((const float4*)(W + (size_t)n * NFEAT))[k4];
    float* base = &sWs[n * WSTRIDE + (k4 << 1)];
    *(float2*)(base)      = make_float2(w.x, w.y);   // half 0 (K=4k4, 4k4+1)
    *(float2*)(base + 64) = make_float2(w.z, w.w);   // half 1 (K=4k4+2, 4k4+3)
  }
  wait_asynccnt0();
  __syncthreads();

  const int wave = tid >> 5;
  const int lane = tid & 31;
  const int half = lane >> 4;      // 0: lanes 0-15 (K=0,1), 1: lanes 16-31 (K=2,3)
  const int l16  = lane & 15;
  const int r0   = wave * 16;
  const int aRow = r0 + l16;       // A-matrix M index for this lane

  // Preload all 32 A fragment pairs (64 VGPRs), reused across all 8 col tiles.
  v2f af[32];
#pragma unroll
  for (int k4 = 0; k4 < 32; ++k4) {
    const int kk = k4 * 4 + half * 2;
    af[k4].x = sX[aRow * LDS_STRIDE + kk];
    af[k4].y = sX[aRow * LDS_STRIDE + kk + 1];
  }
  __builtin_amdgcn_sched_barrier(0);

  for (int ct = 0; ct < 8; ++ct) {
    const int c0   = ct * 16;
    const int bCol = c0 + l16;     // B-matrix N index for this lane (B[k,n] = W[n,k])

    // This lane's 32 B pairs = 64 contiguous floats = 16 x ds_load_b128.
    const float4* bbase = (const float4*)&sWs[bCol * WSTRIDE + half * 64];
    float4 bf4[16];
#pragma unroll
    for (int q = 0; q < 16; ++q) bf4[q] = bbase[q];
    __builtin_amdgcn_sched_barrier(0);

    // Two accumulator chains hide WMMA pass-through latency.
    v8f acc0 = {}, acc1 = {};
#pragma unroll
    for (int k4 = 0; k4 < 32; k4 += 2) {
      v2f b0; b0.x = bf4[k4 >> 1].x; b0.y = bf4[k4 >> 1].y;
      v2f b1; b1.x = bf4[k4 >> 1].z; b1.y = bf4[k4 >> 1].w;
      acc0 = __builtin_amdgcn_wmma_f32_16x16x4_f32(
          false, af[k4],     false, b0, (short)0, acc0, false, false);
      acc1 = __builtin_amdgcn_wmma_f32_16x16x4_f32(
          false, af[k4 + 1], false, b1, (short)0, acc1, false, false);
    }
    v8f acc = acc0 + acc1;
    __builtin_amdgcn_sched_barrier(0);

    // D layout: VGPR r -> M = r + 8*half (lanes 0-15: M=r, lanes 16-31: M=r+8), N = l16
#pragma unroll
    for (int r = 0; r < 8; ++r) {
      int m = row_base + r0 + r + half * 8;
      int n = c0 + l16;
      if (m < N) xw[(size_t)m * NFEAT + n] = acc[r];
    }
  }
}

// ---------------- edge scatter: agg[dst] += xw[src] * dinv[src]*dinv[dst] ----------------
// One wave per edge; each lane handles 4 consecutive features (float4 gather,
// 4 global f32 atomics). xw and agg are both L2-resident (51 MB each < 192 MB L2).
__global__ __launch_bounds__(256) void scatter_kernel(
    const int* __restrict__ src, const int* __restrict__ dst,
    const float* __restrict__ dinv, const float* __restrict__ xw,
    float* __restrict__ agg, int E) {
  int e = blockIdx.x * 8 + (threadIdx.x >> 5);
  if (e >= E) return;
  int lane = threadIdx.x & 31;
  int s = src[e], d = dst[e];
  float norm = dinv[s] * dinv[d];
  float4 v = ((const float4*)(xw + (size_t)s * NFEAT))[lane];
  float* o = agg + (size_t)d * NFEAT + lane * 4;
  atomicAdd(o + 0, v.x * norm);
  atomicAdd(o + 1, v.y * norm);
  atomicAdd(o + 2, v.z * norm);
  atomicAdd(o + 3, v.w * norm);
}

// ---------------- finalize: add self-loop msg + bias, ReLU, then linear+sigmoid ----------------
// One wave per node. h written in place (h region of d_out doubles as agg buffer).
__global__ __launch_bounds__(256) void finalize_kernel(
    const float* __restrict__ xw, const float* __restrict__ dinv,
    const float* __restrict__ b_conv, const float* __restrict__ W_lin,
    const float* __restrict__ b_lin, float* __restrict__ h,
    float* __restrict__ out, int N) {
  int n = blockIdx.x * 8 + (threadIdx.x >> 5);
  if (n >= N) return;
  int lane = threadIdx.x & 31;
  float di = dinv[n];
  float sn = di * di;                       // self-loop norm
  float4 a  = ((const float4*)(h  + (size_t)n * NFEAT))[lane];
  float4 xv = ((const float4*)(xw + (size_t)n * NFEAT))[lane];
  float4 bb = ((const float4*)b_conv)[lane];
  float4 wl = ((const float4*)W_lin)[lane];
  float h0 = fmaxf(fmaf(xv.x, sn, a.x) + bb.x, 0.0f);
  float h1 = fmaxf(fmaf(xv.y, sn, a.y) + bb.y, 0.0f);
  float h2 = fmaxf(fmaf(xv.z, sn, a.z) + bb.z, 0.0f);
  float h3 = fmaxf(fmaf(xv.w, sn, a.w) + bb.w, 0.0f);
  float4 hv = {h0, h1, h2, h3};
  ((float4*)(h + (size_t)n * NFEAT))[lane] = hv;
  float p = h0 * wl.x + h1 * wl.y + h2 * wl.z + h3 * wl.w;
#pragma unroll
  for (int off = 16; off > 0; off >>= 1) p += __shfl_xor(p, off, 32);
  if (lane == 0) out[n] = 1.0f / (1.0f + __expf(-(p + b_lin[0])));
}

extern "C" void kernel_launch(void* const* d_in, const int* in_sizes, int n_in,
                              void* d_out, int out_size, void* d_ws, size_t ws_size,
                              hipStream_t stream) {
  const float* x      = (const float*)d_in[0];
  const int*   ei     = (const int*)d_in[1];
  const float* W_conv = (const float*)d_in[2];
  const float* b_conv = (const float*)d_in[3];
  const float* W_lin  = (const float*)d_in[4];
  const float* b_lin  = (const float*)d_in[5];

  const int N = in_sizes[0] / NFEAT;   // 100000
  const int E = in_sizes[1] / 2;       // 640000
  const int* src = ei;                 // edge_index[0]
  const int* dst = ei + E;             // edge_index[1]

  // d_out layout: tuple (out [N], h [N*128]) concatenated flat
  float* out = (float*)d_out;
  float* h   = (float*)d_out + N;      // also used as the scatter-add accumulator

  // workspace: [deg/dinv: N floats][xw: N*128 floats]  (~51.6 MB)
  float* deg = (float*)d_ws;
  float* xw  = (float*)d_ws + N;

  zero_f32<<<512, 256, 0, stream>>>(deg, (long long)N);
  zero_f32<<<2048, 256, 0, stream>>>(h, (long long)N * NFEAT);
  degree_kernel<<<1024, 256, 0, stream>>>(dst, deg, E);
  dinv_kernel<<<(N + 255) / 256, 256, 0, stream>>>(deg, N);

  const size_t smem_bytes = 2u * 128u * LDS_STRIDE * sizeof(float); // 135168 B (<320KB WGP LDS)
  gemm_xw_wmma<<<(N + 127) / 128, 256, smem_bytes, stream>>>(x, W_conv, xw, N);

  scatter_kernel<<<(E + 7) / 8, 256, 0, stream>>>(src, dst, deg, xw, h, E);
  finalize_kernel<<<(N + 7) / 8, 256, 0, stream>>>(xw, deg, b_conv, W_lin, b_lin, h, out, N);
}